// EdgeGCN_6279242187120
// MI455X (gfx1250) — compile-verified
//
#include <hip/hip_runtime.h>
#include <stdint.h>

typedef __attribute__((ext_vector_type(2))) float v2f;
typedef __attribute__((ext_vector_type(8))) float v8f;

#define N_NODES  50000
#define N_EDGES  800000
#define NODE_DIM 128
#define STATE    64
#define OUT_DIM  32
#define ETYPES   8
#define OUT_W    (ETYPES * OUT_DIM)   // 256

// ---------------------------------------------------------------- utilities
__global__ void EdgeGCN_zero_kernel(float* __restrict__ p, long long n) {
    long long i = (long long)blockIdx.x * blockDim.x + threadIdx.x;
    if (i < n) p[i] = 0.0f;
}

// decode one-hot edge_attr (E x 8) -> int type per edge
__global__ void EdgeGCN_etype_kernel(const float* __restrict__ ea,
                                     int* __restrict__ et) {
    int e = blockIdx.x * blockDim.x + threadIdx.x;
    if (e >= N_EDGES) return;
    const float* a = ea + (long long)e * ETYPES;
    int t = 0;
#pragma unroll
    for (int j = 1; j < ETYPES; ++j)
        if (a[j] > 0.5f) t = j;
    et[e] = t;
}

// per-(type,col) degree count (accumulated into dinv buffer as raw counts)
__global__ void EdgeGCN_deg_kernel(const int* __restrict__ et,
                                   const int* __restrict__ eidx,
                                   float* __restrict__ deg) {
    int e = blockIdx.x * blockDim.x + threadIdx.x;
    if (e >= N_EDGES) return;
    int t = et[e];
    int c = eidx[N_EDGES + e];
    atomicAdd(&deg[(long long)t * N_NODES + c], 1.0f);
}

// deg -> rsqrt(deg + 1) in place
__global__ void EdgeGCN_dinv_kernel(float* __restrict__ d) {
    long long i = (long long)blockIdx.x * blockDim.x + threadIdx.x;
    if (i < (long long)ETYPES * N_NODES) d[i] = rsqrtf(d[i] + 1.0f);
}

// ---------------------------------------------------------------- WMMA GEMM
// C[nrows x S] = A[nrows x K] @ B[K x S], f32, V_WMMA_F32_16X16X4_F32.
// 256 threads = 8 waves; each wave owns one 16x16 C tile. B staged in LDS.
// Row guards: OOB lanes have their A row clamped to 0 (they compute row-0's
// valid dot products), and the C store is guarded per row — no masking math
// needed, and control flow stays uniform so EXEC is all-ones at every WMMA.
template <int K, int S>
__global__ __launch_bounds__(256) void EdgeGCN_gemm_wmma(
        const float* __restrict__ A, const float* __restrict__ B,
        float* __restrict__ C, int nrows) {
    __shared__ float sB[K * S];
    const int tid = threadIdx.x;
    for (int i = tid; i < K * S; i += 256) sB[i] = B[i];
    __syncthreads();

    const int lane = tid & 31;
    const int w    = tid >> 5;
    constexpr int colTiles     = S / 16;          // 4 (S=64) or 2 (S=32)
    constexpr int rowsPerBlock = 16 * (8 / colTiles);
    const int row0 = blockIdx.x * rowsPerBlock + (w / colTiles) * 16;
    const int col0 = (w % colTiles) * 16;
    const int lo = lane & 15;
    const int hi = lane >> 4;

    // A fragment row for this lane (ISA 16x4 layout: M = lane%16,
    // Kpair = 2*(lane/16)); clamp OOB rows to row 0.
    const int m = row0 + lo;
    const long long arow = (long long)((m < nrows) ? m : 0) * K;

    v8f c = {};
#pragma unroll
    for (int kb = 0; kb < K; kb += 4) {
        v2f a = *(const v2f*)(A + arow + kb + 2 * hi);
        v2f b;
        b.x = sB[(kb + 2 * hi)     * S + col0 + lo];
        b.y = sB[(kb + 2 * hi + 1) * S + col0 + lo];
        c = __builtin_amdgcn_wmma_f32_16x16x4_f32(
                /*neg_a=*/false, a, /*neg_b=*/false, b,
                /*c_mod=*/(short)0, c, /*reuse_a=*/false, /*reuse_b=*/false);
    }

    // D layout: VGPR v -> row (v + 8*hi), col = lane%16
#pragma unroll
    for (int v = 0; v < 8; ++v) {
        int mr = row0 + v + 8 * hi;
        if (mr < nrows) C[(long long)mr * S + col0 + lo] = c[v];
    }
}

// ---------------------------------------------------------------- scatters
// Layer 1: 8 threads per edge, 8 floats each (STATE = 64)
__global__ void EdgeGCN_scatter1_kernel(const int* __restrict__ et,
                                        const int* __restrict__ eidx,
                                        const float* __restrict__ dinv_t,
                                        const float* __restrict__ h1,
                                        float* __restrict__ agg, int t) {
    long long tid = (long long)blockIdx.x * blockDim.x + threadIdx.x;
    if (tid >= (long long)N_EDGES * 8) return;
    int e  = (int)(tid >> 3);
    int s0 = (int)(tid & 7) * 8;
    if (et[e] != t) return;
    int r = eidx[e];
    int c = eidx[N_EDGES + e];
    float norm = dinv_t[r] * dinv_t[c];
    const float4* hp = (const float4*)(h1 + (long long)r * STATE + s0);
    float4 ha = hp[0], hb = hp[1];
    float* ap = agg + (long long)c * STATE + s0;
    atomicAdd(ap + 0, ha.x * norm); atomicAdd(ap + 1, ha.y * norm);
    atomicAdd(ap + 2, ha.z * norm); atomicAdd(ap + 3, ha.w * norm);
    atomicAdd(ap + 4, hb.x * norm); atomicAdd(ap + 5, hb.y * norm);
    atomicAdd(ap + 6, hb.z * norm); atomicAdd(ap + 7, hb.w * norm);
}

// agg = relu(agg + h1 * dinv^2 + b1[t]) in place
__global__ void EdgeGCN_combine1_kernel(const float* __restrict__ dinv_t,
                                        const float* __restrict__ h1,
                                        const float* __restrict__ b1t,
                                        float* __restrict__ agg) {
    long long i = (long long)blockIdx.x * blockDim.x + threadIdx.x;
    if (i >= (long long)N_NODES * STATE) return;
    int n = (int)(i >> 6);
    int s = (int)(i & (STATE - 1));
    float di = dinv_t[n];
    float v  = agg[i] + h1[i] * di * di + b1t[s];
    agg[i]   = fmaxf(v, 0.0f);
}

// Layer 2: 8 threads per edge, 4 floats each (OUT_DIM = 32), into d_out slice
__global__ void EdgeGCN_scatter2_kernel(const int* __restrict__ et,
                                        const int* __restrict__ eidx,
                                        const float* __restrict__ dinv_t,
                                        const float* __restrict__ h2,
                                        float* __restrict__ out, int t) {
    long long tid = (long long)blockIdx.x * blockDim.x + threadIdx.x;
    if (tid >= (long long)N_EDGES * 8) return;
    int e  = (int)(tid >> 3);
    int s0 = (int)(tid & 7) * 4;
    if (et[e] != t) return;
    int r = eidx[e];
    int c = eidx[N_EDGES + e];
    float norm = dinv_t[r] * dinv_t[c];
    float4 h = *(const float4*)(h2 + (long long)r * OUT_DIM + s0);
    float* op = out + (long long)c * OUT_W + t * OUT_DIM + s0;
    atomicAdd(op + 0, h.x * norm); atomicAdd(op + 1, h.y * norm);
    atomicAdd(op + 2, h.z * norm); atomicAdd(op + 3, h.w * norm);
}

// out slice += h2 * dinv^2 + b2[t]
__global__ void EdgeGCN_combine2_kernel(const float* __restrict__ dinv_t,
                                        const float* __restrict__ h2,
                                        const float* __restrict__ b2t,
                                        float* __restrict__ out, int t) {
    long long i = (long long)blockIdx.x * blockDim.x + threadIdx.x;
    if (i >= (long long)N_NODES * OUT_DIM) return;
    int n = (int)(i >> 5);
    int s = (int)(i & (OUT_DIM - 1));
    float di = dinv_t[n];
    long long oi = (long long)n * OUT_W + t * OUT_DIM + s;
    out[oi] += h2[i] * di * di + b2t[s];
}

// ---------------------------------------------------------------- launcher
extern "C" void kernel_launch(void* const* d_in, const int* in_sizes, int n_in,
                              void* d_out, int out_size, void* d_ws, size_t ws_size,
                              hipStream_t stream) {
    const float* x   = (const float*)d_in[0];
    const float* ea  = (const float*)d_in[1];
    const float* W1  = (const float*)d_in[2];
    const float* b1  = (const float*)d_in[3];
    const float* W2  = (const float*)d_in[4];
    const float* b2  = (const float*)d_in[5];
    const int*   eix = (const int*)d_in[6];
    float* out = (float*)d_out;

    // workspace layout (256B aligned slices), total ~37 MB
    char* ws = (char*)d_ws;
    auto alignup = [](size_t v) { return (v + 255) & ~(size_t)255; };
    size_t off = 0;
    float* dinv = (float*)(ws + off); off += alignup(sizeof(float) * ETYPES * N_NODES);
    int*   et   = (int*)  (ws + off); off += alignup(sizeof(int) * N_EDGES);
    float* h1   = (float*)(ws + off); off += alignup(sizeof(float) * (size_t)N_NODES * STATE);
    float* agg1 = (float*)(ws + off); off += alignup(sizeof(float) * (size_t)N_NODES * STATE);
    float* h2   = (float*)(ws + off); off += alignup(sizeof(float) * (size_t)N_NODES * OUT_DIM);
    (void)ws_size; (void)in_sizes; (void)n_in; (void)out_size;

    const int TB = 256;
    auto blocksLL = [&](long long n) { return (unsigned)((n + TB - 1) / TB); };

    // 1) init
    EdgeGCN_zero_kernel<<<blocksLL((long long)N_NODES * OUT_W), TB, 0, stream>>>(
        out, (long long)N_NODES * OUT_W);
    EdgeGCN_zero_kernel<<<blocksLL((long long)ETYPES * N_NODES), TB, 0, stream>>>(
        dinv, (long long)ETYPES * N_NODES);
    EdgeGCN_etype_kernel<<<blocksLL(N_EDGES), TB, 0, stream>>>(ea, et);
    EdgeGCN_deg_kernel<<<blocksLL(N_EDGES), TB, 0, stream>>>(et, eix, dinv);
    EdgeGCN_dinv_kernel<<<blocksLL((long long)ETYPES * N_NODES), TB, 0, stream>>>(dinv);

    // 2) per-edge-type two-layer GCN, buffers reused across types
    for (int t = 0; t < ETYPES; ++t) {
        const float* dinv_t = dinv + (long long)t * N_NODES;

        // GEMM1: h1 = x @ W1[t]   (N x 128 @ 128 x 64)
        {
            constexpr int rowsPerBlock = 16 * (8 / (STATE / 16)); // 32
            unsigned nb = (N_NODES + rowsPerBlock - 1) / rowsPerBlock;
            EdgeGCN_gemm_wmma<NODE_DIM, STATE><<<nb, TB, 0, stream>>>(
                x, W1 + (long long)t * NODE_DIM * STATE, h1, N_NODES);
        }

        EdgeGCN_zero_kernel<<<blocksLL((long long)N_NODES * STATE), TB, 0, stream>>>(
            agg1, (long long)N_NODES * STATE);
        EdgeGCN_scatter1_kernel<<<blocksLL((long long)N_EDGES * 8), TB, 0, stream>>>(
            et, eix, dinv_t, h1, agg1, t);
        EdgeGCN_combine1_kernel<<<blocksLL((long long)N_NODES * STATE), TB, 0, stream>>>(
            dinv_t, h1, b1 + (long long)t * STATE, agg1);

        // GEMM2: h2 = relu_h @ W2[t]   (N x 64 @ 64 x 32)
        {
            constexpr int rowsPerBlock = 16 * (8 / (OUT_DIM / 16)); // 64
            unsigned nb = (N_NODES + rowsPerBlock - 1) / rowsPerBlock;
            EdgeGCN_gemm_wmma<STATE, OUT_DIM><<<nb, TB, 0, stream>>>(
                agg1, W2 + (long long)t * STATE * OUT_DIM, h2, N_NODES);
        }

        EdgeGCN_scatter2_kernel<<<blocksLL((long long)N_EDGES * 8), TB, 0, stream>>>(
            et, eix, dinv_t, h2, out, t);
        EdgeGCN_combine2_kernel<<<blocksLL((long long)N_NODES * OUT_DIM), TB, 0, stream>>>(
            dinv_t, h2, b2 + (long long)t * OUT_DIM, out, t);
    }
}